// Model_34093450396597
// MI455X (gfx1250) — compile-verified
//
#include <hip/hip_runtime.h>

#define Bn 32
#define Tn 2048
#define Nn 48
#define Cc 6
#define KMAX 95
#define TM (Tn - 1)
#define RS 2052  // padded LDS row stride (floats): 2048 + 4, 16B-aligned

typedef __attribute__((ext_vector_type(2))) float v2f;
typedef __attribute__((ext_vector_type(4))) float v4f;
typedef __attribute__((ext_vector_type(8))) float v8f;
typedef __attribute__((ext_vector_type(4))) int v4i;

#if defined(__has_builtin)
#if __has_builtin(__builtin_amdgcn_global_load_async_to_lds_b128) && \
    __has_builtin(__builtin_amdgcn_s_wait_asynccnt)
#define HAVE_ASYNC_LDS 1
#endif
#endif

static __device__ __forceinline__ v8f wmma4(v2f a, v2f b, v8f c) {
  // V_WMMA_F32_16X16X4_F32: D = A(16x4 f32) * B(4x16 f32) + C(16x16 f32)
  return __builtin_amdgcn_wmma_f32_16x16x4_f32(false, a, false, b, (short)0, c,
                                               false, false);
}

// ---------------- K1: per-batch Gram matrix A_b = x_b^T x_b / sqrt(T) --------
__global__ void __launch_bounds__(32) k_gram(const float* __restrict__ x,
                                             float* __restrict__ Araw) {
  int b = blockIdx.x;
  int lane = threadIdx.x;
  int m = lane & 15, hi = lane >> 4;
  v8f zero = {0.f, 0.f, 0.f, 0.f, 0.f, 0.f, 0.f, 0.f};
  v8f acc[3][3];
  for (int r = 0; r < 3; ++r)
    for (int s = 0; s < 3; ++s) acc[r][s] = zero;
  const float* xb = x + (size_t)b * Tn * Nn;
  for (int t = 0; t < Tn; t += 4) {
    int tb = t + (hi << 1);
    v2f f[3];
#pragma unroll
    for (int r = 0; r < 3; ++r) {
      f[r].x = xb[(size_t)tb * Nn + 16 * r + m];
      f[r].y = xb[(size_t)(tb + 1) * Nn + 16 * r + m];
    }
#pragma unroll
    for (int r = 0; r < 3; ++r)
#pragma unroll
      for (int s = 0; s < 3; ++s) acc[r][s] = wmma4(f[r], f[s], acc[r][s]);
  }
  const float scale = 0.022097086912079608f;  // 1/sqrt(2048)
  for (int r = 0; r < 3; ++r)
    for (int s = 0; s < 3; ++s)
#pragma unroll
      for (int rr = 0; rr < 8; ++rr) {
        int nrow = 16 * r + rr + (hi << 3);
        int mcol = 16 * s + m;
        Araw[((size_t)b * Nn + nrow) * Nn + mcol] = acc[r][s][rr] * scale;
      }
}

// ---------------- K2: leaky_relu -> row softmax -> mean over batch -----------
__global__ void k_rowsoftmax(const float* __restrict__ Araw,
                             float* __restrict__ A) {
  int n = blockIdx.x, tid = threadIdx.x;  // 64 threads
  __shared__ float sh[64];
  float accm = 0.0f;
  for (int b = 0; b < Bn; ++b) {
    float v = -3.0e38f;
    if (tid < Nn) {
      float a = Araw[((size_t)b * Nn + n) * Nn + tid];
      v = (a >= 0.0f) ? a : 0.2f * a;
    }
    sh[tid] = v;
    __syncthreads();
    for (int off = 32; off > 0; off >>= 1) {
      if (tid < off) sh[tid] = fmaxf(sh[tid], sh[tid + off]);
      __syncthreads();
    }
    float mx = sh[0];
    __syncthreads();
    float e = (tid < Nn) ? __expf(v - mx) : 0.0f;
    sh[tid] = e;
    __syncthreads();
    for (int off = 32; off > 0; off >>= 1) {
      if (tid < off) sh[tid] += sh[tid + off];
      __syncthreads();
    }
    float sum = sh[0];
    __syncthreads();
    accm += e / sum;
  }
  if (tid < Nn) A[n * Nn + tid] = accm * (1.0f / Bn);
}

// ---------------- K3: Laplacian + Chebyshev stack mul_L[6,48,48] -------------
__global__ void k_cheb(const float* __restrict__ A, float* __restrict__ mulL) {
  __shared__ float dsh[Nn];
  __shared__ float Lp[Nn * Nn], Cm2[Nn * Nn], Cm1[Nn * Nn], Cn[Nn * Nn];
  int tid = threadIdx.x;  // 256
  if (tid < Nn) {
    float s = 0.0f;
    for (int mm = 0; mm < Nn; ++mm) s += A[tid * Nn + mm];
    dsh[tid] = 1.0f / sqrtf(s + 1e-5f);
  }
  __syncthreads();
  for (int idx = tid; idx < Nn * Nn; idx += 256) {
    int nr = idx / Nn, mc = idx % Nn;
    float iden = (nr == mc) ? 1.0f : 0.0f;
    float lap = iden - dsh[nr] * A[idx] * dsh[mc];
    Lp[idx] = lap;
    Cm2[idx] = iden;
    Cm1[idx] = lap;
    mulL[idx] = iden;
    mulL[Nn * Nn + idx] = lap;
  }
  __syncthreads();
  for (int kk = 2; kk < Cc; ++kk) {
    for (int idx = tid; idx < Nn * Nn; idx += 256) {
      int nr = idx / Nn, mc = idx % Nn;
      float s = 0.0f;
      for (int r = 0; r < Nn; ++r) s += Lp[nr * Nn + r] * Cm1[r * Nn + mc];
      Cn[idx] = 2.0f * s - Cm2[idx];
    }
    __syncthreads();
    for (int idx = tid; idx < Nn * Nn; idx += 256) {
      mulL[kk * Nn * Nn + idx] = Cn[idx];
      Cm2[idx] = Cm1[idx];
      Cm1[idx] = Cn[idx];
    }
    __syncthreads();
  }
}

// ---------------- K4: fused GX einsum + QKV projections ----------------------
// q,k stored [N,B,C,T] (row-per-channel, feeds LDS rows in k_corr)
// v stored [N,B,T,C] (channel-interleaved, feeds gather in k_agg)
__global__ void k_gxqkv(const float* __restrict__ x,
                        const float* __restrict__ mulL,
                        const float* __restrict__ Wq,
                        const float* __restrict__ Wk,
                        const float* __restrict__ Wv, float* __restrict__ q,
                        float* __restrict__ k, float* __restrict__ v) {
  int tid = threadIdx.x;
  int b = blockIdx.y, n = blockIdx.z;
  int t = blockIdx.x * 256 + tid;
  __shared__ float Lsh[Cc * Nn];
  __shared__ float Wsh[3 * Cc * Cc];
  for (int idx = tid; idx < Cc * Nn; idx += 256) {
    int kk = idx / Nn, mm = idx % Nn;
    Lsh[idx] = mulL[(kk * Nn + n) * Nn + mm];
  }
  if (tid < Cc * Cc) {
    Wsh[tid] = Wq[tid];
    Wsh[36 + tid] = Wk[tid];
    Wsh[72 + tid] = Wv[tid];
  }
  __syncthreads();
  float xr[Nn];
  const float* xp = x + ((size_t)b * Tn + t) * Nn;
#pragma unroll
  for (int mm = 0; mm < Nn; ++mm) xr[mm] = xp[mm];
  float g[Cc];
#pragma unroll
  for (int kk = 0; kk < Cc; ++kk) {
    float s = 0.0f;
    for (int mm = 0; mm < Nn; ++mm) s += Lsh[kk * Nn + mm] * xr[mm];
    g[kk] = s;
  }
  size_t nb = (size_t)(n * Bn + b);
  size_t obv = (nb * Tn + t) * Cc;
#pragma unroll
  for (int c = 0; c < Cc; ++c) {
    float sq = 0.f, sk = 0.f, sv = 0.f;
#pragma unroll
    for (int kk = 0; kk < Cc; ++kk) {
      sq += g[kk] * Wsh[kk * Cc + c];
      sk += g[kk] * Wsh[36 + kk * Cc + c];
      sv += g[kk] * Wsh[72 + kk * Cc + c];
    }
    q[(nb * Cc + c) * Tn + t] = sq;
    k[(nb * Cc + c) * Tn + t] = sk;
    v[obv + c] = sv;
  }
}

// ---------------- K5: circular autocorrelation via f32 WMMA ------------------
// mean_val[n,b,tau] = (1/C) sum_{t,c} q[t,c] * k[(t-tau) mod T, c]
// tau = 16a+i; A-frag = Toeplitz tile of K, B-frag = shifted tile of Q.
// LDS rows are [c][t] with a +4 pad; element [2048] duplicates [0] so the
// (t, t+1) operand pair is always adjacent -> single ds_load_2addr_b32.
__global__ void __launch_bounds__(256) k_corr(const float* __restrict__ q,
                                              const float* __restrict__ kq,
                                              float* __restrict__ mv) {
  __shared__ float Qs[Cc * RS];
  __shared__ float Ks[Cc * RS];
  int tid = threadIdx.x;
  int b = blockIdx.x, n = blockIdx.y;
  size_t base = (size_t)(n * Bn + b) * (Tn * Cc);
  const float* gq = q + base;
  const float* gk = kq + base;
#if defined(HAVE_ASYNC_LDS)
  for (int i = tid; i < (Tn * Cc) / 4; i += 256) {
    int c = i >> 9, pos = (i & 511) << 2;
    __builtin_amdgcn_global_load_async_to_lds_b128(
        (__attribute__((address_space(1))) v4i*)(gq + (i << 2)),
        (__attribute__((address_space(3))) v4i*)(Qs + c * RS + pos), 0, 0);
    __builtin_amdgcn_global_load_async_to_lds_b128(
        (__attribute__((address_space(1))) v4i*)(gk + (i << 2)),
        (__attribute__((address_space(3))) v4i*)(Ks + c * RS + pos), 0, 0);
  }
  __builtin_amdgcn_s_wait_asynccnt(0);
  __syncthreads();
#else
  for (int i = tid; i < (Tn * Cc) / 4; i += 256) {
    int c = i >> 9, pos = (i & 511) << 2;
    *(v4f*)(Qs + c * RS + pos) = *(const v4f*)(gq + (i << 2));
    *(v4f*)(Ks + c * RS + pos) = *(const v4f*)(gk + (i << 2));
  }
  __syncthreads();
#endif
  if (tid < 2 * Cc) {  // circular-wrap pad: row[2048] = row[0]
    float* p = (tid < Cc ? Qs : Ks) + (tid % Cc) * RS;
    p[Tn] = p[0];
  }
  __syncthreads();
  int lane = tid & 31, wid = tid >> 5;
  int m = lane & 15, hi = lane >> 4;
  int a = (wid << 4) + m;  // output-tile column index (tau tile id)
  v8f acc = {0.f, 0.f, 0.f, 0.f, 0.f, 0.f, 0.f, 0.f};
  for (int w = 0; w < 128; ++w) {
    int ka0 = (w << 4) + (hi << 1) - m;    // K time index, + v + 4*js
    int kb0 = ((a + w) << 4) + (hi << 1);  // Q time index, + v + 4*js
#pragma unroll
    for (int c = 0; c < Cc; ++c) {
      const float* Krow = Ks + c * RS;
      const float* Qrow = Qs + c * RS;
#pragma unroll
      for (int js = 0; js < 4; ++js) {
        int ia = (ka0 + (js << 2)) & TM;
        int ib = (kb0 + (js << 2)) & TM;
        v2f af, bf;
        af.x = Krow[ia];
        af.y = Krow[ia + 1];  // pad makes this valid at ia==2047
        bf.x = Qrow[ib];
        bf.y = Qrow[ib + 1];
        acc = wmma4(af, bf, acc);
      }
    }
  }
  size_t ob = (size_t)(n * Bn + b) * Tn;
#pragma unroll
  for (int r = 0; r < 8; ++r) {
    int i = r + (hi << 3);
    int tau = (a << 4) + i;
    mv[ob + tau] = acc[r] * (1.0f / 6.0f);
  }
}

// ---------------- K6: mean over batch + top-7 delays per node ----------------
__global__ void k_topk(const float* __restrict__ mv, int* __restrict__ idxo) {
  int n = blockIdx.x, tid = threadIdx.x;  // 256
  __shared__ float gsh[Tn];
  __shared__ float rv[256];
  __shared__ int ri[256];
  for (int t = tid; t < Tn; t += 256) {
    float s = 0.0f;
    for (int b = 0; b < Bn; ++b) s += mv[(size_t)(n * Bn + b) * Tn + t];
    gsh[t] = s;
  }
  __syncthreads();
  for (int j = 0; j < 7; ++j) {
    float bv = -3.0e38f;
    int bi = 0;
    for (int t = tid; t < Tn; t += 256) {
      float vv = gsh[t];
      if (vv > bv) { bv = vv; bi = t; }
    }
    rv[tid] = bv;
    ri[tid] = bi;
    __syncthreads();
    for (int off = 128; off > 0; off >>= 1) {
      if (tid < off) {
        if (rv[tid + off] > rv[tid] ||
            (rv[tid + off] == rv[tid] && ri[tid + off] < ri[tid])) {
          rv[tid] = rv[tid + off];
          ri[tid] = ri[tid + off];
        }
      }
      __syncthreads();
    }
    if (tid == 0) {
      idxo[n * 7 + j] = ri[0];
      gsh[ri[0]] = -3.0e38f;
    }
    __syncthreads();
  }
}

// ---------------- K7: per-(n,b) softmax weights over 7 delays ----------------
__global__ void k_wts(const float* __restrict__ mv, const int* __restrict__ idxo,
                      float* __restrict__ wts) {
  int n = blockIdx.x, b = threadIdx.x;  // 32 threads
  float vals[7];
  float mx = -3.0e38f;
  for (int j = 0; j < 7; ++j) {
    vals[j] = mv[(size_t)(n * Bn + b) * Tn + idxo[n * 7 + j]];
    mx = fmaxf(mx, vals[j]);
  }
  float s = 0.0f;
  for (int j = 0; j < 7; ++j) {
    vals[j] = __expf(vals[j] - mx);
    s += vals[j];
  }
  float inv = 1.0f / s;
  for (int j = 0; j < 7; ++j) wts[(n * Bn + b) * 7 + j] = vals[j] * inv;
}

// ---------------- K8: delay aggregation (rolled gather of v) -----------------
__global__ void k_agg(const float* __restrict__ v, const int* __restrict__ idxo,
                      const float* __restrict__ wts, float* __restrict__ agg) {
  int tid = threadIdx.x;
  int b = blockIdx.y, n = blockIdx.z;
  int t = blockIdx.x * 256 + tid;
  int id7[7];
  float w7[7];
#pragma unroll
  for (int j = 0; j < 7; ++j) {
    id7[j] = idxo[n * 7 + j];
    w7[j] = wts[(n * Bn + b) * 7 + j];
  }
  size_t nb = (size_t)(n * Bn + b) * Tn;
  float s[Cc];
#pragma unroll
  for (int c = 0; c < Cc; ++c) s[c] = 0.0f;
#pragma unroll
  for (int j = 0; j < 7; ++j) {
    const float* vp = v + (nb + ((t + id7[j]) & TM)) * Cc;
    float wj = w7[j];
#pragma unroll
    for (int c = 0; c < Cc; ++c) s[c] += vp[c] * wj;
  }
  float* op = agg + (nb + t) * Cc;
#pragma unroll
  for (int c = 0; c < Cc; ++c) op[c] = s[c];
}

// ---------------- K9: per-node conv (ksz=2i+1) + bn + relu -> xs[N,B,C,T] ----
__global__ void k_conv(const float* __restrict__ agg,
                       const float* __restrict__ cw,
                       const float* __restrict__ cb,
                       const float* __restrict__ bg,
                       const float* __restrict__ bb, float* __restrict__ xs) {
  int tid = threadIdx.x;
  int b = blockIdx.y, i = blockIdx.z;
  int t0 = blockIdx.x * 256;
  int ksz = 2 * i + 1;
  __shared__ float insh[(256 + 2 * 47) * Cc];
  __shared__ float wsh[Cc * Cc * KMAX];
  __shared__ float cbs[Cc], bgs[Cc], bbs[Cc];
  const float rs = 0.9999950000374996f;  // 1/sqrt(1+1e-5)
  if (tid < Cc) {
    cbs[tid] = cb[i * Cc + tid];
    bgs[tid] = bg[i * Cc + tid] * rs;
    bbs[tid] = bb[i * Cc + tid];
  }
  int cnt = Cc * Cc * ksz;
  for (int idx = tid; idx < cnt; idx += 256) {
    int d = idx % ksz;
    int rem = idx / ksz;
    int ci = rem % Cc;
    int co = rem / Cc;
    wsh[idx] = cw[(((size_t)i * Cc + co) * Cc + ci) * KMAX + d];
  }
  int wlen = (256 + 2 * i) * Cc;
  size_t nb = (size_t)(i * Bn + b) * Tn;
  for (int idx = tid; idx < wlen; idx += 256) {
    int tt = t0 - i + idx / Cc;
    int c = idx % Cc;
    insh[idx] = (tt >= 0 && tt < Tn) ? agg[(nb + tt) * Cc + c] : 0.0f;
  }
  __syncthreads();
  int t = t0 + tid;
  for (int co = 0; co < Cc; ++co) {
    float s = cbs[co];
    const float* wrow = wsh + co * Cc * ksz;
    for (int d = 0; d < ksz; ++d) {
      const float* ip = insh + (tid + d) * Cc;
#pragma unroll
      for (int ci = 0; ci < Cc; ++ci) s += wrow[ci * ksz + d] * ip[ci];
    }
    float y = s * bgs[co] + bbs[co];
    xs[((size_t)(i * Bn + b) * Cc + co) * Tn + t] = fmaxf(y, 0.0f);
  }
}

// ---------------- K10: z[b,c] = sum over i, mean over t of xs ------------------------
__global__ void k_zred(const float* __restrict__ xs, float* __restrict__ z) {
  int bc = blockIdx.x;
  int b = bc / Cc, c = bc % Cc;
  int tid = threadIdx.x;
  __shared__ float sh[256];
  float s = 0.0f;
  for (int idx = tid; idx < Nn * Tn; idx += 256) {
    int i = idx / Tn;
    int t = idx % Tn;
    s += xs[((size_t)(i * Bn + b) * Cc + c) * Tn + t];
  }
  sh[tid] = s;
  __syncthreads();
  for (int off = 128; off > 0; off >>= 1) {
    if (tid < off) sh[tid] += sh[tid + off];
    __syncthreads();
  }
  if (tid == 0) z[b * Cc + c] = sh[0] * (1.0f / Tn);
}

// ---------------- K11: gating softmax c[b,l] ---------------------------------
__global__ void k_gate(const float* __restrict__ z,
                       const float* __restrict__ bn2g,
                       const float* __restrict__ bn2b,
                       const float* __restrict__ W, float* __restrict__ gate) {
  int b = threadIdx.x;  // 32
  const float rs = 0.9999950000374996f;
  float zz[Cc];
#pragma unroll
  for (int c = 0; c < Cc; ++c) {
    float t = z[b * Cc + c] * bn2g[c] * rs + bn2b[c];
    zz[c] = fmaxf(t, 0.0f);
  }
  float sarr[Nn];
  float mx = -3.0e38f;
  for (int l = 0; l < Nn; ++l) {
    float s = 0.0f;
#pragma unroll
    for (int c = 0; c < Cc; ++c) s += zz[c] * W[c * Nn + l];
    sarr[l] = s;
    mx = fmaxf(mx, s);
  }
  float sum = 0.0f;
  for (int l = 0; l < Nn; ++l) {
    sarr[l] = __expf(sarr[l] - mx);
    sum += sarr[l];
  }
  float inv = 1.0f / sum;
  for (int l = 0; l < Nn; ++l) gate[b * Nn + l] = sarr[l] * inv;
}

// ---------------- K12: final gated mix + residual ----------------------------
__global__ void k_final(const float* __restrict__ xs,
                        const float* __restrict__ gate,
                        const float* __restrict__ x, float* __restrict__ out) {
  int id = blockIdx.x * 256 + threadIdx.x;  // B*T*N
  int b = id / (Tn * Nn);
  int rem = id % (Tn * Nn);
  int t = rem / Nn;
  int l = rem % Nn;
  float s = 0.0f;
#pragma unroll
  for (int c = 0; c < Cc; ++c)
    s += xs[((size_t)(l * Bn + b) * Cc + c) * Tn + t];
  out[id] = s * gate[b * Nn + l] * (1.0f / Cc) + x[id];
}

extern "C" void kernel_launch(void* const* d_in, const int* in_sizes, int n_in,
                              void* d_out, int out_size, void* d_ws,
                              size_t ws_size, hipStream_t stream) {
  (void)in_sizes; (void)n_in; (void)out_size; (void)ws_size;
  const float* x = (const float*)d_in[0];
  const float* Wq = (const float*)d_in[1];
  const float* Wk = (const float*)d_in[2];
  const float* Wv = (const float*)d_in[3];
  const float* cw = (const float*)d_in[4];
  const float* cb = (const float*)d_in[5];
  const float* bg = (const float*)d_in[6];
  const float* bb = (const float*)d_in[7];
  const float* b2g = (const float*)d_in[8];
  const float* b2b = (const float*)d_in[9];
  const float* W = (const float*)d_in[10];
  float* out = (float*)d_out;
  char* ws = (char*)d_ws;

  float* Araw = (float*)(ws + 0);        // 32*48*48
  float* A = (float*)(ws + 294912);      // 48*48
  float* mulL = (float*)(ws + 304128);   // 6*48*48
  int* idxo = (int*)(ws + 359424);       // 48*7
  float* wts = (float*)(ws + 360768);    // 48*32*7
  float* z = (float*)(ws + 403776);      // 32*6
  float* gate = (float*)(ws + 404544);   // 32*48
  const size_t big = 75497472;           // N*B*T*C*4 bytes
  float* q = (float*)(ws + 1048576);
  float* k = (float*)(ws + 1048576 + big);
  float* v = (float*)(ws + 1048576 + 2 * big);
  float* mv = (float*)(ws + 1048576 + 3 * big);  // N*B*T
  float* agg = q;  // q dead after k_corr
  float* xs = k;   // k dead after k_corr

  k_gram<<<dim3(Bn), 32, 0, stream>>>(x, Araw);
  k_rowsoftmax<<<dim3(Nn), 64, 0, stream>>>(Araw, A);
  k_cheb<<<1, 256, 0, stream>>>(A, mulL);
  k_gxqkv<<<dim3(Tn / 256, Bn, Nn), 256, 0, stream>>>(x, mulL, Wq, Wk, Wv, q, k,
                                                      v);
  k_corr<<<dim3(Bn, Nn), 256, 0, stream>>>(q, k, mv);
  k_topk<<<dim3(Nn), 256, 0, stream>>>(mv, idxo);
  k_wts<<<dim3(Nn), 32, 0, stream>>>(mv, idxo, wts);
  k_agg<<<dim3(Tn / 256, Bn, Nn), 256, 0, stream>>>(v, idxo, wts, agg);
  k_conv<<<dim3(Tn / 256, Bn, Nn), 256, 0, stream>>>(agg, cw, cb, bg, bb, xs);
  k_zred<<<dim3(Bn * Cc), 256, 0, stream>>>(xs, z);
  k_gate<<<1, 32, 0, stream>>>(z, b2g, b2b, W, gate);
  k_final<<<dim3((Bn * Tn * Nn) / 256), 256, 0, stream>>>(xs, gate, x, out);
}